// ComplexEMA_68315749810465
// MI455X (gfx1250) — compile-verified
//
#include <hip/hip_runtime.h>
#include <math.h>

// ---------------------------------------------------------------------------
// Complex EMA (MEGA-style) on MI455X / gfx1250.
//
// Roofline: mandatory traffic = 128 MB (x in + y out) -> ~5.5us @ 23.3 TB/s.
// Naive VALU scan = ~13 GFLOP fp32 -> ~15x slower than the memory floor.
// So we reformulate the linear recurrence as chunked (T=16) matmuls and run
// all bulk arithmetic on V_WMMA_F32_16X16X4_F32 (fp32 WMMA keeps reference
// precision; matrix-unit throughput makes the FLOPs free vs. HBM).
//
//   y[t] = Sum_{k<=t in chunk} K[t-k] x[k]            (A_d @ X,  WMMA)
//        + Re( g . q^{t'+1} h_chunk_in )              (W_d @ H,  WMMA)
//        + omega * x[t]                               (fused in epilogue)
//   h_out = q^16 (.) h_in + P_d @ x_chunk             (P_d @ X,  WMMA; the
//                                                      128-step/b scan is a
//                                                      tiny wave32 shuffle op)
// ---------------------------------------------------------------------------

#define D_EMBED 2048
#define NDIM    16
#define BB      4
#define LL      2048
#define T       16                       // chunk length == WMMA tile dim
#define CPB     (LL / T)                 // 128 chunks per batch row
#define NCHUNK  (BB * CPB)               // 512 chunk-columns per d
#define NTILE   (NCHUNK / 16)            // 32 column tiles

typedef float v2f __attribute__((ext_vector_type(2)));
typedef float v8f __attribute__((ext_vector_type(8)));

// workspace layout (floats)
#define A_OFF   0                        // [D][16][16]  intra-chunk Toeplitz
#define P_OFF   (D_EMBED * 256)          // [D][32][16]  carry-injection (re;im)
#define W_OFF   (P_OFF + D_EMBED * 512)  // [D][16][32]  state readout
#define QT_OFF  (W_OFF + D_EMBED * 512)  // [D][32]      q^16 (re[16], im[16])

__device__ __forceinline__ float sigmoidf_(float v) {
  return 1.0f / (1.0f + expf(-v));
}

// ---------------------------------------------------------------------------
// Coefficient kernel: one thread per d. Builds A, P, W, q^16 in d_ws.
// ---------------------------------------------------------------------------
__global__ void ema_coeff_kernel(const float* __restrict__ alpha,
                                 const float* __restrict__ delta,
                                 const float* __restrict__ theta,
                                 const float* __restrict__ gamma,
                                 float* __restrict__ ws) {
  const int d = blockIdx.x * blockDim.x + threadIdx.x;
  if (d >= D_EMBED) return;

  const float scale = rsqrtf((float)NDIM);
  const float sth = sigmoidf_(theta[d]) * (6.283185307179586f / (float)NDIM);

  float* Ad = ws + A_OFF + d * 256;
  float* Pd = ws + P_OFF + d * 512;
  float* Wd = ws + W_OFF + d * 512;
  float* Qd = ws + QT_OFF + d * 32;

  float K[T];
#pragma unroll
  for (int t = 0; t < T; ++t) K[t] = 0.0f;

  for (int n = 0; n < NDIM; ++n) {
    const float p  = sigmoidf_(alpha[d * NDIM + n]);
    const float dd = sigmoidf_(delta[d * NDIM + n]);
    const float phi = (float)(n + 1) * sth;
    const float mag = 1.0f - p * dd;
    const float qr = mag * cosf(phi);
    const float qi = mag * sinf(phi);
    const float gr = gamma[(d * NDIM + n) * 2 + 0] * scale;
    const float gi = gamma[(d * NDIM + n) * 2 + 1] * scale;

    float cr = 1.0f, ci = 0.0f;  // q^tau
    for (int tau = 0; tau <= T; ++tau) {
      if (tau < T) {
        K[tau] += p * (gr * cr - gi * ci);          // p * Re(g q^tau)
        Pd[n * 16 + (15 - tau)]        = p * cr;    // p * Re(q^{15-k})
        Pd[(16 + n) * 16 + (15 - tau)] = p * ci;    // p * Im(q^{15-k})
      }
      if (tau >= 1) {
        const int t = tau - 1;
        Wd[t * 32 + n]      =  gr * cr - gi * ci;   //  Re(g q^{t+1})
        Wd[t * 32 + 16 + n] = -(gr * ci + gi * cr); // -Im(g q^{t+1})
      }
      if (tau == T) { Qd[n] = cr; Qd[16 + n] = ci; }
      const float ncr = cr * qr - ci * qi;
      const float nci = cr * qi + ci * qr;
      cr = ncr; ci = nci;
    }
  }
  for (int t = 0; t < T; ++t)
    for (int k = 0; k < T; ++k)
      Ad[t * 16 + k] = (t >= k) ? K[t - k] : 0.0f;
}

// ---------------------------------------------------------------------------
// Main kernel: one workgroup (8 wave32) per d.
// Phase A: WMMA  y_intra = A@X  and  S = P@X  -> S into LDS
// Phase B: waves 0..3 run the 128-step carry scan (in-place S -> H in LDS)
// Phase C: WMMA  y += W@H, fused omega*x residual, contiguous b128 stores
// ---------------------------------------------------------------------------
__global__ __launch_bounds__(256) void ema_scan_kernel(
    const float* __restrict__ x, const float* __restrict__ omega,
    const float* __restrict__ ws, float* __restrict__ y) {
  const int d    = blockIdx.x;
  const int lane = threadIdx.x & 31;
  const int wv   = threadIdx.x >> 5;  // 0..7
  const int lp   = lane & 15;
  const int hi   = lane >> 4;
  const int koff = hi * 2;            // K-pair select inside a K=4 step

  __shared__ float Hlds[NCHUNK * 32]; // 64 KB: S, then (in-place) chunk-in H

  const float* Ad = ws + A_OFF + (size_t)d * 256;
  const float* Pd = ws + P_OFF + (size_t)d * 512;
  const float* Wd = ws + W_OFF + (size_t)d * 512;
  const float* Qd = ws + QT_OFF + (size_t)d * 32;

  // Coefficient A-fragments (uniform across column tiles) -- hoisted.
  v2f fA[4], fP0[4], fP1[4];
#pragma unroll
  for (int kk = 0; kk < 4; ++kk) {
    const int k0 = kk * 4 + koff;
    fA[kk]  = *(const v2f*)(Ad + lp * 16 + k0);
    fP0[kk] = *(const v2f*)(Pd + lp * 16 + k0);
    fP1[kk] = *(const v2f*)(Pd + (16 + lp) * 16 + k0);
  }

  v8f accY[4];
#pragma unroll
  for (int i = 0; i < 4; ++i) accY[i] = {};

  // ---------------- Phase A ----------------
#pragma unroll
  for (int i = 0; i < 4; ++i) {
    const int tile = wv * 4 + i;
    const int col  = tile * 16 + lp;     // chunk-column this lane feeds
    const int b    = col >> 7;
    const int c    = col & 127;
    const float* xcol = x + ((size_t)b * D_EMBED + d) * LL + (size_t)c * 16;

    v8f sRe = {}, sIm = {};
#pragma unroll
    for (int kk = 0; kk < 4; ++kk) {
      const int k0 = kk * 4 + koff;
      const v2f bf = *(const v2f*)(xcol + k0);   // X[k][col], contiguous b64
      accY[i] = __builtin_amdgcn_wmma_f32_16x16x4_f32(
          false, fA[kk], false, bf, (short)0, accY[i], false, false);
      sRe = __builtin_amdgcn_wmma_f32_16x16x4_f32(
          false, fP0[kk], false, bf, (short)0, sRe, false, false);
      sIm = __builtin_amdgcn_wmma_f32_16x16x4_f32(
          false, fP1[kk], false, bf, (short)0, sIm, false, false);
    }
    // D layout: VGPR v <-> row v + 8*hi, col = lp. 32B-aligned LDS stores.
    *(v8f*)(&Hlds[col * 32 + 0 * 16 + 8 * hi]) = sRe;
    *(v8f*)(&Hlds[col * 32 + 1 * 16 + 8 * hi]) = sIm;
  }

  __syncthreads();

  // ---------------- Phase B: carry scan (waves 0..3 <-> batch rows) -------
  if (wv < 4) {
    const int b = wv;
    const float qtr = Qd[lp];           // Re(q^16)
    const float qti = Qd[16 + lp];      // Im(q^16)
    const float sgn = (hi == 0) ? -qti : qti;  // re' = qtr*re - qti*im
                                               // im' = qtr*im + qti*re
    float h = 0.0f;                     // lane<16: Re(h_n); lane>=16: Im(h_n)
    for (int c = 0; c < CPB; ++c) {
      const int base = (b * CPB + c) * 32 + lane;
      const float s = Hlds[base];
      Hlds[base] = h;                   // publish chunk-incoming state
      const float partner = __shfl_xor(h, 16, 32);
      h = qtr * h + sgn * partner + s;
    }
  }

  __syncthreads();

  // ---------------- Phase C: y += W @ H, residual, store ------------------
  const float om = omega[d];
#pragma unroll
  for (int i = 0; i < 4; ++i) {
    const int tile = wv * 4 + i;
    const int col  = tile * 16 + lp;
    v8f acc = accY[i];
#pragma unroll
    for (int kk = 0; kk < 8; ++kk) {    // K = 32 (re;im state components)
      const int k0 = kk * 4 + koff;
      const v2f af = *(const v2f*)(Wd + lp * 32 + k0);
      const v2f bf = *(const v2f*)(&Hlds[col * 32 + k0]);
      acc = __builtin_amdgcn_wmma_f32_16x16x4_f32(
          false, af, false, bf, (short)0, acc, false, false);
    }
    const int b = col >> 7;
    const int c = col & 127;
    const size_t rowbase =
        ((size_t)b * D_EMBED + d) * LL + (size_t)c * 16 + 8 * hi;
    const float* xr = x + rowbase;      // 8 contiguous floats per lane
    float* yr = y + rowbase;
#pragma unroll
    for (int v = 0; v < 8; ++v) yr[v] = acc[v] + om * xr[v];
  }
}

extern "C" void kernel_launch(void* const* d_in, const int* in_sizes, int n_in,
                              void* d_out, int out_size, void* d_ws,
                              size_t ws_size, hipStream_t stream) {
  (void)in_sizes; (void)n_in; (void)out_size; (void)ws_size;
  const float* x     = (const float*)d_in[0];
  const float* alpha = (const float*)d_in[1];
  const float* delta = (const float*)d_in[2];
  const float* theta = (const float*)d_in[3];
  const float* gamma = (const float*)d_in[4];
  const float* omega = (const float*)d_in[5];
  float* y  = (float*)d_out;
  float* ws = (float*)d_ws;   // needs (2048*1312)*4 = ~10.75 MB

  ema_coeff_kernel<<<D_EMBED / 256, 256, 0, stream>>>(alpha, delta, theta,
                                                      gamma, ws);
  ema_scan_kernel<<<D_EMBED, 256, 0, stream>>>(x, omega, ws, y);
}